// Backward_14328010900205
// MI455X (gfx1250) — compile-verified
//
#include <hip/hip_runtime.h>

// Backward warp (grid_sample bilinear, zeros padding) + validity mask.
// Memory-bound gather kernel: AI ~1 FLOP/byte, floor ~18us at 23.3 TB/s.
// WMMA intentionally not used (no matrix structure); CDNA5 levers used:
// wave32-coalesced gathers, per-pixel weight amortization across the 196
// strided channels, and non-temporal output stores to keep the 192MB L2
// reserved for the gather stream.

namespace {
constexpr int kB = 8, kC = 196, kH = 128, kW = 256;
constexpr int kHW  = kH * kW;        // 32768
constexpr int kCHW = kC * kHW;       // 6,422,528
constexpr int kChunk   = 49;         // channels per thread (196 / 4)
constexpr int kThreads = 256;        // 8 wave32 waves per block
}

__global__ __launch_bounds__(kThreads)
void warp_bilinear_kernel(const float* __restrict__ inp,
                          const float* __restrict__ flow,
                          float* __restrict__ out) {
  const int pix = blockIdx.x * kThreads + threadIdx.x;   // 0 .. B*H*W-1
  const int w = pix & (kW - 1);
  const int h = (pix >> 8) & (kH - 1);
  const int b = pix >> 15;
  const int c0 = blockIdx.y * kChunk;

  // Flow for this pixel (tiny; L2-resident across the 4 channel chunks).
  const float fx = flow[((b * 2 + 0) * kH + h) * kW + w];
  const float fy = flow[((b * 2 + 1) * kH + h) * kW + w];

  // Mirror the reference arithmetic: gx = linspace + flow/((W-1)/2), then
  // x = (gx+1)*0.5*(W-1).
  const float gx = (-1.0f + (float)w * (2.0f / (kW - 1))) + fx * (2.0f / (kW - 1));
  const float gy = (-1.0f + (float)h * (2.0f / (kH - 1))) + fy * (2.0f / (kH - 1));
  const float x = (gx + 1.0f) * 0.5f * (float)(kW - 1);
  const float y = (gy + 1.0f) * 0.5f * (float)(kH - 1);

  const float x0f = floorf(x), y0f = floorf(y);
  const float x1f = x0f + 1.0f, y1f = y0f + 1.0f;
  const float wx1 = x - x0f, wx0 = 1.0f - wx1;
  const float wy1 = y - y0f, wy0 = 1.0f - wy1;

  const bool vx0 = (x0f >= 0.0f) && (x0f <= (float)(kW - 1));
  const bool vx1 = (x1f >= 0.0f) && (x1f <= (float)(kW - 1));
  const bool vy0 = (y0f >= 0.0f) && (y0f <= (float)(kH - 1));
  const bool vy1 = (y1f >= 0.0f) && (y1f <= (float)(kH - 1));

  // Fold per-corner validity into the bilinear weights (zeros padding).
  const float W00 = wx0 * wy0 * ((vx0 && vy0) ? 1.0f : 0.0f);
  const float W10 = wx1 * wy0 * ((vx1 && vy0) ? 1.0f : 0.0f);
  const float W01 = wx0 * wy1 * ((vx0 && vy1) ? 1.0f : 0.0f);
  const float W11 = wx1 * wy1 * ((vx1 && vy1) ? 1.0f : 0.0f);
  // Warped "ones" channel == sum of valid weights; mask = (val > 0.999).
  const float maskval = (W00 + W10) + (W01 + W11);

  // Clamped integer corner offsets (shared across all channels).
  const int x0 = (int)fminf(fmaxf(x0f, 0.0f), (float)(kW - 1));
  const int x1 = (int)fminf(fmaxf(x1f, 0.0f), (float)(kW - 1));
  const int y0 = (int)fminf(fmaxf(y0f, 0.0f), (float)(kH - 1));
  const int y1 = (int)fminf(fmaxf(y1f, 0.0f), (float)(kH - 1));
  const int o00 = y0 * kW + x0, o10 = y0 * kW + x1;
  const int o01 = y1 * kW + x0, o11 = y1 * kW + x1;

  const float* __restrict__ p  = inp + (size_t)b * kCHW + (size_t)c0 * kHW;
  float* __restrict__       po = out + (size_t)b * kCHW + (size_t)c0 * kHW
                                     + (size_t)(h * kW + w);

  if (maskval > 0.999f) {
    // Stream this pixel's channel slice: 4 gathers + 3 FMAs + 1 NT store per
    // channel; unroll x7 gives 28 loads in flight per thread.
#pragma unroll 7
    for (int c = 0; c < kChunk; ++c) {
      const float* __restrict__ q = p + (size_t)c * kHW;
      float r =        q[o00] * W00;
      r = fmaf(q[o10], W10, r);
      r = fmaf(q[o01], W01, r);
      r = fmaf(q[o11], W11, r);
      __builtin_nontemporal_store(r, po + (size_t)c * kHW);
    }
  } else {
    // Masked-out pixel: skip all gathers, emit zeros (output is poisoned).
#pragma unroll 7
    for (int c = 0; c < kChunk; ++c) {
      __builtin_nontemporal_store(0.0f, po + (size_t)c * kHW);
    }
  }
}

extern "C" void kernel_launch(void* const* d_in, const int* in_sizes, int n_in,
                              void* d_out, int out_size, void* d_ws, size_t ws_size,
                              hipStream_t stream) {
  (void)in_sizes; (void)n_in; (void)out_size; (void)d_ws; (void)ws_size;
  const float* inp  = (const float*)d_in[0];   // (8,196,128,256) fp32
  const float* flow = (const float*)d_in[1];   // (8,2,128,256)  fp32
  float* out = (float*)d_out;                  // (8,196,128,256) fp32

  dim3 grid((kB * kH * kW) / kThreads,         // 1024 pixel blocks
            kC / kChunk);                      // 4 channel chunks
  warp_bilinear_kernel<<<grid, kThreads, 0, stream>>>(inp, flow, out);
}